// TransformerBlock_52243982188860
// MI455X (gfx1250) — compile-verified
//
#include <hip/hip_runtime.h>
#include <math.h>

// ---------------------------------------------------------------------------
// MI455X (gfx1250) decode-step transformer block. Bandwidth-bound fp32:
// ~1.9 GB of weights/cache per call -> ~82us floor at 23.3 TB/s.
// All M=8 GEMVs use V_WMMA_F32_16X16X4_F32 (wave32, 16x16 C tile).
// Activation buffers are padded to 16 rows (rows 8-15 zeroed once per call)
// so the GEMV inner loop has NO predication -> EXEC stays all-ones, no
// saveexec churn around the WMMAs. Weights stream with NT cache hints.
// ---------------------------------------------------------------------------

typedef float v2f __attribute__((ext_vector_type(2)));
typedef float v4f __attribute__((ext_vector_type(4)));
typedef float v8f __attribute__((ext_vector_type(8)));

#define DIM      8192
#define QKV_N    10240
#define HEAD_DIM 128
#define N_KV     8
#define N_REP    8
#define KV       4096
#define NB       8          // batch (tokens)
#define FD       2048
#define FE       1024
#define NE       8          // experts

__device__ __forceinline__ float gelu_exact(float x) {
  return 0.5f * x * (1.0f + erff(x * 0.70710678118654752440f));
}

// ---------------------------------------------------------------------------
// WMMA GEMV: Ypart[z][8][N] = A[16][K-chunk z] x W.  One wave per 16-col tile.
// LAYKN=0: W is NK, element(k,n) = W[n*ldw + k]   (all the w.T matmuls)
// LAYKN=1: W is KN, element(k,n) = W[k*ldw + n]   (wqkv)
// A must have 16 rows with rows 8..15 zero-filled.
// WMMA K-slot mapping per 8-k step: lanes 0-15 carry K-slots{0,1}=k,k+1;
// lanes 16-31 carry slots{2,3}=k+4,k+5 (2nd WMMA: k+2,k+3 / k+6,k+7).
// A and B use the same permutation of K, so the product is unchanged.
// ---------------------------------------------------------------------------
template <int LAYKN>
__global__ void __launch_bounds__(32)
gemv8_wmma(const float* __restrict__ A, int lda, long astride,
           const float* __restrict__ W, int ldw, long wstride,
           float* __restrict__ Y, int ldy, long ystride, long zstride,
           int kchunk, int N)
{
  const int lane = threadIdx.x & 31;
  const int hi   = lane >> 4;        // 0: K-slots 0,1   1: K-slots 2,3
  const int nl   = lane & 15;
  A += (long)blockIdx.y * astride;
  W += (long)blockIdx.y * wstride;
  Y += (long)blockIdx.y * ystride + (long)blockIdx.z * zstride;

  const int n0  = blockIdx.x * 16;
  const int k0  = blockIdx.z * kchunk;
  const int myN = n0 + nl;

  const float* ap = A + (long)nl * lda + k0 + 4 * hi;
  v8f c = {};

  if (LAYKN) {
    const float* wp = W + myN + (long)(k0 + 4 * hi) * ldw;
    const long ldl = (long)ldw;
#pragma unroll 4
    for (int k = 0; k < kchunk; k += 8) {
      v4f a4 = *(const v4f*)ap;
      v2f b1 = { __builtin_nontemporal_load(wp),
                 __builtin_nontemporal_load(wp + ldl) };
      v2f b2 = { __builtin_nontemporal_load(wp + 2 * ldl),
                 __builtin_nontemporal_load(wp + 3 * ldl) };
      __builtin_prefetch(wp + 16 * ldl, 0, 1);          // global_prefetch_b8
      v2f a1 = {a4.x, a4.y}, a2 = {a4.z, a4.w};
      c = __builtin_amdgcn_wmma_f32_16x16x4_f32(false, a1, false, b1,
                                                (short)0, c, false, false);
      c = __builtin_amdgcn_wmma_f32_16x16x4_f32(false, a2, false, b2,
                                                (short)0, c, false, false);
      ap += 8; wp += 8 * ldl;
    }
  } else {
    const float* wp = W + (long)myN * ldw + k0 + 4 * hi;
#pragma unroll 4
    for (int k = 0; k < kchunk; k += 8) {
      v4f a4 = *(const v4f*)ap;
      v4f w4 = __builtin_nontemporal_load((const v4f*)wp);
      __builtin_prefetch(wp + 512, 0, 1);               // global_prefetch_b8
      v2f a1 = {a4.x, a4.y}, a2 = {a4.z, a4.w};
      v2f b1 = {w4.x, w4.y}, b2 = {w4.z, w4.w};
      c = __builtin_amdgcn_wmma_f32_16x16x4_f32(false, a1, false, b1,
                                                (short)0, c, false, false);
      c = __builtin_amdgcn_wmma_f32_16x16x4_f32(false, a2, false, b2,
                                                (short)0, c, false, false);
      ap += 8; wp += 8;
    }
  }

  // C/D layout: VGPR j, lanes 0-15 -> row M=j, col N=lane. Tokens are M=0..7.
  if (lane < 16) {
#pragma unroll
    for (int j = 0; j < 8; ++j)
      Y[(long)j * ldy + myN] = c[j];
  }
}

// ---------------------------------------------------------------------------
// Combine K-split partials: Y[i] = (R?R[i]:0) + sum_z part[z*zstride + i]
// ---------------------------------------------------------------------------
__global__ void __launch_bounds__(256)
combine_kernel(const float* __restrict__ part, int S, long zstride,
               const float* __restrict__ R, float* __restrict__ Y, int n)
{
  int i = blockIdx.x * 256 + threadIdx.x;
  if (i < n) {
    float s = R ? R[i] : 0.f;
    for (int z = 0; z < S; ++z) s += part[(long)z * zstride + i];
    Y[i] = s;
  }
}

// ---------------------------------------------------------------------------
// Zero the pad rows (8..15) of 16-row activation tiles: nb blocks of
// blockstride floats, pad at padoff, padlen floats each.
// ---------------------------------------------------------------------------
__global__ void __launch_bounds__(256)
zero_pad_kernel(float* __restrict__ p, long blockstride, long padoff,
                int padlen, int nb)
{
  int i = blockIdx.x * 256 + threadIdx.x;
  if (i < padlen * nb) {
    int b = i / padlen, r = i - b * padlen;
    p[(long)b * blockstride + padoff + r] = 0.f;
  }
}

// ---------------------------------------------------------------------------
// RMSNorm: one block per row (writes rows 0..7 only; pads stay zero).
// ---------------------------------------------------------------------------
__global__ void __launch_bounds__(256)
rmsnorm_kernel(const float* __restrict__ x, const float* __restrict__ w,
               float* __restrict__ y, int dim)
{
  const int row = blockIdx.x;
  const float* xr = x + (long)row * dim;
  float ss = 0.f;
  for (int i = threadIdx.x; i < dim; i += 256) { float v = xr[i]; ss += v * v; }
#pragma unroll
  for (int off = 16; off; off >>= 1) ss += __shfl_xor(ss, off, 32);
  __shared__ float red[8];
  __shared__ float scale_s;
  if ((threadIdx.x & 31) == 0) red[threadIdx.x >> 5] = ss;
  __syncthreads();
  if (threadIdx.x == 0) {
    float t = 0.f;
#pragma unroll
    for (int i = 0; i < 8; ++i) t += red[i];
    scale_s = rsqrtf(t / (float)dim + 1e-5f);
  }
  __syncthreads();
  const float sc = scale_s;
  for (int i = threadIdx.x; i < dim; i += 256)
    y[(long)row * dim + i] = xr[i] * sc * w[i];
}

// ---------------------------------------------------------------------------
// RoPE: rotate q (in place in qkv buffer) and new k; copy new v.
// ---------------------------------------------------------------------------
__global__ void __launch_bounds__(64)
rope_kernel(float* __restrict__ qkv, const float* __restrict__ fcos,
            const float* __restrict__ fsin,
            float* __restrict__ knew, float* __restrict__ vnew)
{
  const int b = blockIdx.x >> 3, h = blockIdx.x & 7;
  const int d2 = threadIdx.x;                  // 0..63
  const float c = fcos[d2], s = fsin[d2];
  const long gbase = (long)b * QKV_N + h * 1280;
#pragma unroll
  for (int r = 0; r < N_REP; ++r) {
    const long qb = gbase + r * HEAD_DIM;
    float re = qkv[qb + 2 * d2], im = qkv[qb + 2 * d2 + 1];
    qkv[qb + 2 * d2]     = re * c - im * s;
    qkv[qb + 2 * d2 + 1] = re * s + im * c;
  }
  const long kb = gbase + 1024;
  const long ob = (long)(b * N_KV + h) * HEAD_DIM;
  {
    float re = qkv[kb + 2 * d2], im = qkv[kb + 2 * d2 + 1];
    knew[ob + 2 * d2]     = re * c - im * s;
    knew[ob + 2 * d2 + 1] = re * s + im * c;
  }
  vnew[ob + 2 * d2]     = qkv[kb + HEAD_DIM + 2 * d2];
  vnew[ob + 2 * d2 + 1] = qkv[kb + HEAD_DIM + 2 * d2 + 1];
}

// ---------------------------------------------------------------------------
// GQA attention, decode. One block (256 thr = 8 waves) per (b, kv-head).
// Scores staged in global scratch; V tiled through LDS; KV cache streamed NT.
// The slot at start_pos uses the freshly computed k/v (caches read-only).
// ---------------------------------------------------------------------------
__global__ void __launch_bounds__(256)
attn_kernel(const float* __restrict__ qkv,
            const float* __restrict__ cache_k, const float* __restrict__ cache_v,
            const float* __restrict__ knew, const float* __restrict__ vnew,
            const float* __restrict__ mask, const int* __restrict__ sp_ptr,
            float* __restrict__ sc_g,     // [64][8][4096] scratch
            float* __restrict__ out)      // [16][8192] (rows 0..7 written)
{
  __shared__ float q_s[N_REP * HEAD_DIM];
  __shared__ float vt[64 * HEAD_DIM];
  __shared__ float invs[N_REP];

  const int b = blockIdx.x >> 3, h = blockIdx.x & 7;
  const int tid = threadIdx.x, lane = tid & 31, wv = tid >> 5;
  const int sp = sp_ptr[0];
  float* sc = sc_g + (long)blockIdx.x * (N_REP * KV);

  for (int i = tid; i < N_REP * HEAD_DIM; i += 256)
    q_s[i] = qkv[(long)b * QKV_N + h * 1280 + i];
  __syncthreads();

  const float scale = 0.08838834764831845f;  // 1/sqrt(128)
  const float* kn = knew + (long)(b * N_KV + h) * HEAD_DIM;
  const float* vn = vnew + (long)(b * N_KV + h) * HEAD_DIM;

  // ---- pass 1: scores ----
  for (int j = wv * 512; j < wv * 512 + 512; ++j) {
    const float* kp = (j == sp) ? kn
                                : cache_k + (((long)b * KV + j) * N_KV + h) * HEAD_DIM;
    v4f k4 = __builtin_nontemporal_load((const v4f*)(kp + lane * 4));
    float p[N_REP];
#pragma unroll
    for (int r = 0; r < N_REP; ++r) {
      const float* qp = q_s + r * HEAD_DIM + lane * 4;
      p[r] = k4.x * qp[0] + k4.y * qp[1] + k4.z * qp[2] + k4.w * qp[3];
    }
#pragma unroll
    for (int off = 16; off; off >>= 1)
#pragma unroll
      for (int r = 0; r < N_REP; ++r) p[r] += __shfl_xor(p[r], off, 32);
    if (lane == 0) {
      const float m = mask[j];
#pragma unroll
      for (int r = 0; r < N_REP; ++r) sc[(long)r * KV + j] = p[r] * scale + m;
    }
  }
  __syncthreads();

  // ---- pass 2: softmax, wave wv handles rep wv ----
  {
    const int r = wv;
    float mx = -3.4e38f;
    for (int j = lane; j < KV; j += 32) mx = fmaxf(mx, sc[(long)r * KV + j]);
#pragma unroll
    for (int off = 16; off; off >>= 1) mx = fmaxf(mx, __shfl_xor(mx, off, 32));
    float s = 0.f;
    for (int j = lane; j < KV; j += 32) {
      float e = __expf(sc[(long)r * KV + j] - mx);
      sc[(long)r * KV + j] = e;
      s += e;
    }
#pragma unroll
    for (int off = 16; off; off >>= 1) s += __shfl_xor(s, off, 32);
    if (lane == 0) invs[r] = 1.0f / s;
  }
  __syncthreads();

  // ---- pass 3: out[r][d] = sum_j p * v ----
  const int r = tid >> 5;
  const int d = (tid & 31) * 4;
  v4f acc = {};
  for (int jt = 0; jt < KV / 64; ++jt) {
    for (int i = tid; i < 64 * HEAD_DIM / 4; i += 256) {   // 2048 float4
      const int jj = i >> 5, col = (i & 31) * 4;
      const int j = jt * 64 + jj;
      const float* vp = (j == sp) ? vn
                                  : cache_v + (((long)b * KV + j) * N_KV + h) * HEAD_DIM;
      *(v4f*)(vt + jj * HEAD_DIM + col) =
          __builtin_nontemporal_load((const v4f*)(vp + col));
    }
    __syncthreads();
#pragma unroll 8
    for (int jj = 0; jj < 64; ++jj) {
      const float p = sc[(long)r * KV + jt * 64 + jj];
      const float* vp = vt + jj * HEAD_DIM + d;
      acc.x += p * vp[0]; acc.y += p * vp[1];
      acc.z += p * vp[2]; acc.w += p * vp[3];
    }
    __syncthreads();
  }
  const float iv = invs[r];
  const long ob = (long)b * DIM + h * (N_REP * HEAD_DIM) + r * HEAD_DIM + d;
  out[ob + 0] = acc.x * iv; out[ob + 1] = acc.y * iv;
  out[ob + 2] = acc.z * iv; out[ob + 3] = acc.w * iv;
}

// ---------------------------------------------------------------------------
// Elementwise: g = gelu(a) * b   (dense FFN; writes flat 8*FD region)
// ---------------------------------------------------------------------------
__global__ void __launch_bounds__(256)
gelu_mul_kernel(const float* __restrict__ a, const float* __restrict__ b,
                float* __restrict__ o, int n)
{
  int i = blockIdx.x * 256 + threadIdx.x;
  if (i < n) o[i] = gelu_exact(a[i]) * b[i];
}

// MoE: gs[e][t][f] = weight[t][e]*gelu(h1)*h3, into 16-row-padded layout
__global__ void __launch_bounds__(256)
gelu_mul_scale_kernel(const float* __restrict__ a, const float* __restrict__ b,
                      const float* __restrict__ wts, float* __restrict__ o, int n)
{
  int i = blockIdx.x * 256 + threadIdx.x;
  if (i < n) {
    const int e = i >> 13;          // / (8*FE)
    const int t = (i >> 10) & 7;    // / FE % 8
    const int f = i & (FE - 1);
    o[(long)e * (16 * FE) + t * FE + f] = wts[t * NE + e] * gelu_exact(a[i]) * b[i];
  }
}

// ---------------------------------------------------------------------------
// Gate logits: 64 blocks = (t,e); one wave each, lane-strided dot.
// ---------------------------------------------------------------------------
__global__ void __launch_bounds__(32)
gate_dot_kernel(const float* __restrict__ fin, const float* __restrict__ gw,
                float* __restrict__ logits)
{
  const int t = blockIdx.x >> 3, e = blockIdx.x & 7;
  const int lane = threadIdx.x;
  const float* a = fin + (long)t * DIM;
  const float* w = gw + (long)e * DIM;
  float s = 0.f;
  for (int k = lane * 4; k < DIM; k += 128) {
    v4f av = *(const v4f*)(a + k);
    v4f wv = *(const v4f*)(w + k);
    s += av.x * wv.x + av.y * wv.y + av.z * wv.z + av.w * wv.w;
  }
#pragma unroll
  for (int off = 16; off; off >>= 1) s += __shfl_xor(s, off, 32);
  if (lane == 0) logits[t * NE + e] = s;
}

// ---------------------------------------------------------------------------
// Gate: softmax over 8 experts, keep top-2 probabilities (no renorm).
// ---------------------------------------------------------------------------
__global__ void __launch_bounds__(32)
gate_topk_kernel(const float* __restrict__ logits, float* __restrict__ weights)
{
  const int t = threadIdx.x;
  if (t < NB) {
    float l[NE];
    float mx = -3.4e38f;
#pragma unroll
    for (int e = 0; e < NE; ++e) { l[e] = logits[t * NE + e]; mx = fmaxf(mx, l[e]); }
    float s = 0.f;
#pragma unroll
    for (int e = 0; e < NE; ++e) { l[e] = __expf(l[e] - mx); s += l[e]; }
    const float is = 1.0f / s;
#pragma unroll
    for (int e = 0; e < NE; ++e) l[e] *= is;
    int i1 = 0;
#pragma unroll
    for (int e = 1; e < NE; ++e) if (l[e] > l[i1]) i1 = e;
    int i2 = (i1 == 0) ? 1 : 0;
#pragma unroll
    for (int e = 0; e < NE; ++e) if (e != i1 && l[e] > l[i2]) i2 = e;
#pragma unroll
    for (int e = 0; e < NE; ++e)
      weights[t * NE + e] = (e == i1 || e == i2) ? l[e] : 0.f;
  }
}

// ---------------------------------------------------------------------------
// out = h + ff + sum_e ye[e]   (deterministic: no float atomics)
// ---------------------------------------------------------------------------
__global__ void __launch_bounds__(256)
final_add_kernel(const float* __restrict__ hb, const float* __restrict__ ff,
                 const float* __restrict__ ye, float* __restrict__ out, int n)
{
  int i = blockIdx.x * 256 + threadIdx.x;
  if (i < n) {
    float s = hb[i] + ff[i];
#pragma unroll
    for (int e = 0; e < NE; ++e) s += ye[(long)e * n + i];
    out[i] = s;
  }
}

// ---------------------------------------------------------------------------
extern "C" void kernel_launch(void* const* d_in, const int* in_sizes, int n_in,
                              void* d_out, int out_size, void* d_ws, size_t ws_size,
                              hipStream_t stream)
{
  (void)in_sizes; (void)n_in; (void)out_size; (void)ws_size;
  const float* x       = (const float*)d_in[0];
  const float* mask    = (const float*)d_in[1];
  const float* fcos    = (const float*)d_in[2];
  const float* fsin    = (const float*)d_in[3];
  const float* cache_k = (const float*)d_in[4];
  const float* cache_v = (const float*)d_in[5];
  const float* wqkv    = (const float*)d_in[6];
  const float* wo      = (const float*)d_in[7];
  const float* attn_nw = (const float*)d_in[8];
  const float* ffn_nw  = (const float*)d_in[9];
  const float* w1      = (const float*)d_in[10];
  const float* w2      = (const float*)d_in[11];
  const float* w3      = (const float*)d_in[12];
  const float* gate_w  = (const float*)d_in[13];
  const float* mw1     = (const float*)d_in[14];
  const float* mw2     = (const float*)d_in[15];
  const float* mw3     = (const float*)d_in[16];
  const int*   sp      = (const int*)d_in[17];
  float* out = (float*)d_out;
  float* ws  = (float*)d_ws;

  // workspace layout (floats); A-inputs padded to 16 rows
  float* an      = ws;                        // 16*8192
  float* qkv     = an      + 16 * DIM;        // 8*10240
  float* attn_o  = qkv     + NB * QKV_N;      // 16*8192
  float* h_buf   = attn_o  + 16 * DIM;        // 8*8192
  float* fin     = h_buf   + NB * DIM;        // 16*8192
  float* ff1     = fin     + 16 * DIM;        // 8*2048
  float* ff3     = ff1     + NB * FD;         // 8*2048
  float* gbuf    = ff3     + NB * FD;         // 16*2048
  float* ff_out  = gbuf    + 16 * FD;         // 8*8192
  float* logits  = ff_out  + NB * DIM;        // 64
  float* weights = logits  + 64;              // 64
  float* knew    = weights + 64;              // 8*8*128
  float* vnew    = knew    + NB * N_KV * HEAD_DIM;
  float* h1b     = vnew    + NB * N_KV * HEAD_DIM;  // E*8*1024
  float* h3b     = h1b     + NE * NB * FE;
  float* gsb     = h3b     + NE * NB * FE;          // E*16*1024 (padded)
  float* yeb     = gsb     + NE * 16 * FE;          // E*8*8192
  float* part    = yeb     + (long)NE * NB * DIM;   // split-partial pool (4*8*10240)
  float* sc_g    = part    + 4 * NB * QKV_N;        // 64*8*4096

  // 0. zero the pad rows of all WMMA A-operand buffers (every call)
  zero_pad_kernel<<<256, 256, 0, stream>>>(an,     0, (long)NB * DIM, NB * DIM, 1);
  zero_pad_kernel<<<256, 256, 0, stream>>>(attn_o, 0, (long)NB * DIM, NB * DIM, 1);
  zero_pad_kernel<<<256, 256, 0, stream>>>(fin,    0, (long)NB * DIM, NB * DIM, 1);
  zero_pad_kernel<<<64,  256, 0, stream>>>(gbuf,   0, (long)NB * FD,  NB * FD,  1);
  zero_pad_kernel<<<256, 256, 0, stream>>>(gsb, 16 * FE, (long)NB * FE, NB * FE, NE);

  // 1. attention RMSNorm
  rmsnorm_kernel<<<NB, 256, 0, stream>>>(x, attn_nw, an, DIM);

  // 2. qkv = an @ wqkv (KN layout), K split x4 -> 2560 waves
  gemv8_wmma<1><<<dim3(QKV_N / 16, 1, 4), 32, 0, stream>>>(
      an, DIM, 0, wqkv, QKV_N, 0, part, QKV_N, 0, (long)NB * QKV_N,
      DIM / 4, QKV_N);
  combine_kernel<<<(NB * QKV_N + 255) / 256, 256, 0, stream>>>(
      part, 4, (long)NB * QKV_N, nullptr, qkv, NB * QKV_N);

  // 3. RoPE (q in place) + new k/v
  rope_kernel<<<NB * N_KV, 64, 0, stream>>>(qkv, fcos, fsin, knew, vnew);

  // 4. GQA attention
  attn_kernel<<<NB * N_KV, 256, 0, stream>>>(qkv, cache_k, cache_v, knew, vnew,
                                             mask, sp, sc_g, attn_o);

  // 5. h = x + attn_o @ wo.T (NK), K split x4, residual fused into combine
  gemv8_wmma<0><<<dim3(DIM / 16, 1, 4), 32, 0, stream>>>(
      attn_o, DIM, 0, wo, DIM, 0, part, DIM, 0, (long)NB * DIM, DIM / 4, DIM);
  combine_kernel<<<(NB * DIM + 255) / 256, 256, 0, stream>>>(
      part, 4, (long)NB * DIM, x, h_buf, NB * DIM);

  // 6. FFN RMSNorm
  rmsnorm_kernel<<<NB, 256, 0, stream>>>(h_buf, ffn_nw, fin, DIM);

  // 7. dense FFN (w1/w3 K split x8, w2 K split x4)
  gemv8_wmma<0><<<dim3(FD / 16, 1, 8), 32, 0, stream>>>(
      fin, DIM, 0, w1, DIM, 0, part, FD, 0, (long)NB * FD, DIM / 8, FD);
  combine_kernel<<<(NB * FD + 255) / 256, 256, 0, stream>>>(
      part, 8, (long)NB * FD, nullptr, ff1, NB * FD);
  gemv8_wmma<0><<<dim3(FD / 16, 1, 8), 32, 0, stream>>>(
      fin, DIM, 0, w3, DIM, 0, part, FD, 0, (long)NB * FD, DIM / 8, FD);
  combine_kernel<<<(NB * FD + 255) / 256, 256, 0, stream>>>(
      part, 8, (long)NB * FD, nullptr, ff3, NB * FD);
  gelu_mul_kernel<<<(NB * FD + 255) / 256, 256, 0, stream>>>(ff1, ff3, gbuf, NB * FD);
  gemv8_wmma<0><<<dim3(DIM / 16, 1, 4), 32, 0, stream>>>(
      gbuf, FD, 0, w2, FD, 0, part, DIM, 0, (long)NB * DIM, FD / 4, DIM);
  combine_kernel<<<(NB * DIM + 255) / 256, 256, 0, stream>>>(
      part, 4, (long)NB * DIM, nullptr, ff_out, NB * DIM);

  // 8. gate + top-2
  gate_dot_kernel<<<NB * NE, 32, 0, stream>>>(fin, gate_w, logits);
  gate_topk_kernel<<<1, 32, 0, stream>>>(logits, weights);

  // 9. MoE up-projections (grid.y = expert, K split x4), dense as in reference
  gemv8_wmma<0><<<dim3(FE / 16, NE, 4), 32, 0, stream>>>(
      fin, DIM, 0, mw1, DIM, (long)FE * DIM, part, FE, (long)NB * FE,
      (long)NE * NB * FE, DIM / 4, FE);
  combine_kernel<<<(NE * NB * FE + 255) / 256, 256, 0, stream>>>(
      part, 4, (long)NE * NB * FE, nullptr, h1b, NE * NB * FE);
  gemv8_wmma<0><<<dim3(FE / 16, NE, 4), 32, 0, stream>>>(
      fin, DIM, 0, mw3, DIM, (long)FE * DIM, part, FE, (long)NB * FE,
      (long)NE * NB * FE, DIM / 4, FE);
  combine_kernel<<<(NE * NB * FE + 255) / 256, 256, 0, stream>>>(
      part, 4, (long)NE * NB * FE, nullptr, h3b, NE * NB * FE);

  // 10. gs[e] = weight[t][e]*gelu(h1)*h3 into padded [E][16][FE]
  gelu_mul_scale_kernel<<<(NE * NB * FE + 255) / 256, 256, 0, stream>>>(
      h1b, h3b, weights, gsb, NE * NB * FE);

  // 11. MoE down-projection (4096 blocks already; no split)
  gemv8_wmma<0><<<dim3(DIM / 16, NE, 1), 32, 0, stream>>>(
      gsb, FE, (long)16 * FE, mw2, FE, (long)DIM * FE,
      yeb, DIM, (long)NB * DIM, 0, FE, DIM);

  // 12. out = h + ff + sum_e ye
  final_add_kernel<<<(NB * DIM + 255) / 256, 256, 0, stream>>>(
      h_buf, ff_out, yeb, out, NB * DIM);
}